// RetroOnlyModel_3204045603544
// MI455X (gfx1250) — compile-verified
//
#include <hip/hip_runtime.h>
#include <hip/hip_bf16.h>

#define H 64
#define L_SEQ 2048
#define CHUNK 64
#define NCHUNK (L_SEQ / CHUNK)

typedef float v2f __attribute__((ext_vector_type(2)));
typedef float v4f __attribute__((ext_vector_type(4)));
typedef float v8f __attribute__((ext_vector_type(8)));

struct Smem {
  float h0[64][68];    // h0, then z (in place)
  float ff[64][132];   // FFN hidden [64x128]
  float h[64][68];     // post-layernorm
  float k[64][68];     // normalized keys for current chunk
  float v[64][68];     // values for current chunk
  float m[64][68];     // M dumped here once, for final readout only
  int   x[64];
  float b1[128];
  float b2[64], lng[64], lnb[64], bo[64];
  float q[64];
  float scores[64];
  float ctx[64];
  float soft[2];
};

// 16x16 f32 output tile via V_WMMA_F32_16X16X4_F32.
// A: LDS activation [*, lda], rows m0..m0+15, K = Klen.
// B: row-major weight W[N][K] in global (computes A @ W^T tile at n0).
__device__ __forceinline__ v8f wmma_tile(const float* __restrict__ A, int lda, int m0,
                                         const float* __restrict__ W, int ldb, int n0,
                                         int Klen, int lane, v8f c) {
  const int mr = m0 + (lane & 15);
  const int nr = n0 + (lane & 15);
  const int ko = (lane >> 4) * 2;
  for (int k0 = 0; k0 < Klen; k0 += 4) {
    const int kb = k0 + ko;
    v2f a; a.x = A[mr * lda + kb];     a.y = A[mr * lda + kb + 1];
    v2f b; b.x = W[nr * ldb + kb];     b.y = W[nr * ldb + kb + 1];
    c = __builtin_amdgcn_wmma_f32_16x16x4_f32(false, a, false, b, (short)0, c, false, false);
  }
  return c;
}

__global__ __launch_bounds__(256) void retro_fused_kernel(
    const int* __restrict__ x, const float* __restrict__ embed,
    const float* __restrict__ w1, const float* __restrict__ b1,
    const float* __restrict__ w2, const float* __restrict__ b2,
    const float* __restrict__ ln_g, const float* __restrict__ ln_b,
    const float* __restrict__ Wk, const float* __restrict__ Wv,
    const float* __restrict__ Wq, const float* __restrict__ Wout,
    const float* __restrict__ bout, float* __restrict__ out) {
  __shared__ Smem s;
  const int b    = blockIdx.x;
  const int tid  = threadIdx.x;
  const int wave = tid >> 5;
  const int lane = tid & 31;
  const int row4 = tid >> 2;   // 0..63 : row this thread owns (4-way split)
  const int part = tid & 3;    // 0..3  : 16-wide column segment
  const int j0   = part * 16;

  // Delta-rule state M[row4][j0 .. j0+15] lives in registers for the whole kernel.
  float mreg[16];
  for (int jj = 0; jj < 16; ++jj) mreg[jj] = 0.f;

  if (tid < 128) s.b1[tid] = b1[tid];
  if (tid < 64) {
    s.b2[tid] = b2[tid]; s.lng[tid] = ln_g[tid];
    s.lnb[tid] = ln_b[tid]; s.bo[tid] = bout[tid];
  }
  __syncthreads();

  for (int ch = 0; ch < NCHUNK; ++ch) {
    const int l0 = ch * CHUNK;
    if (tid < 64) s.x[tid] = x[b * L_SEQ + l0 + tid];
    __syncthreads();   // also fences the previous chunk's scan reads of s.k/s.v

    // ---- embedding gather ----
    {
      const int tok = s.x[row4];
      for (int jj = 0; jj < 16; ++jj)
        s.h0[row4][j0 + jj] = embed[tok * H + j0 + jj];
    }
    __syncthreads();

    // ---- FFN up: ff = relu(h0 @ w1^T + b1)  [64x128], 32 tiles / 8 waves ----
    for (int r = 0; r < 4; ++r) {
      const int id = wave + 8 * r;
      const int mt = id >> 3, nt = id & 7;
      v8f c = {};
      c = wmma_tile(&s.h0[0][0], 68, mt * 16, w1, 64, nt * 16, 64, lane, c);
      const int crow = mt * 16 + (lane >> 4) * 8;
      const int ccol = nt * 16 + (lane & 15);
      for (int vv = 0; vv < 8; ++vv) {
        float val = c[vv] + s.b1[ccol];
        s.ff[crow + vv][ccol] = val > 0.f ? val : 0.f;
      }
    }
    __syncthreads();

    // ---- FFN down + residual: z = h0 + ff @ w2^T + b2 (in place in h0) ----
    for (int r = 0; r < 2; ++r) {
      const int id = wave + 8 * r;
      const int mt = id >> 2, nt = id & 3;
      v8f c = {};
      c = wmma_tile(&s.ff[0][0], 132, mt * 16, w2, 128, nt * 16, 128, lane, c);
      const int crow = mt * 16 + (lane >> 4) * 8;
      const int ccol = nt * 16 + (lane & 15);
      for (int vv = 0; vv < 8; ++vv)
        s.h0[crow + vv][ccol] += c[vv] + s.b2[ccol];
    }
    __syncthreads();

    // ---- layernorm (shuffle reduction over the 4 lanes owning one row) ----
    {
      float sum = 0.f, sq = 0.f;
      for (int jj = 0; jj < 16; ++jj) {
        float z = s.h0[row4][j0 + jj]; sum += z; sq += z * z;
      }
      sum += __shfl_xor(sum, 1, 32); sum += __shfl_xor(sum, 2, 32);
      sq  += __shfl_xor(sq,  1, 32); sq  += __shfl_xor(sq,  2, 32);
      const float mu  = sum * (1.f / 64.f);
      const float var = sq * (1.f / 64.f) - mu * mu;
      const float rs  = rsqrtf(var + 1e-5f);
      for (int jj = 0; jj < 16; ++jj) {
        const int c = j0 + jj;
        s.h[row4][c] = (s.h0[row4][c] - mu) * rs * s.lng[c] + s.lnb[c];
      }
    }
    __syncthreads();

    // ---- K = h @ Wk^T, V = h @ Wv^T  (16 tiles each / 8 waves) ----
    for (int r = 0; r < 2; ++r) {
      const int id = wave + 8 * r;
      const int mt = id >> 2, nt = id & 3;
      const int crow = mt * 16 + (lane >> 4) * 8;
      const int ccol = nt * 16 + (lane & 15);
      v8f ck = {};
      ck = wmma_tile(&s.h[0][0], 68, mt * 16, Wk, 64, nt * 16, 64, lane, ck);
      for (int vv = 0; vv < 8; ++vv) s.k[crow + vv][ccol] = ck[vv];
      v8f cv = {};
      cv = wmma_tile(&s.h[0][0], 68, mt * 16, Wv, 64, nt * 16, 64, lane, cv);
      for (int vv = 0; vv < 8; ++vv) s.v[crow + vv][ccol] = cv[vv];
    }
    __syncthreads();

    // ---- normalize K rows (F.normalize), shuffle reduction ----
    {
      float sq = 0.f;
      for (int jj = 0; jj < 16; ++jj) { float kv = s.k[row4][j0 + jj]; sq += kv * kv; }
      sq += __shfl_xor(sq, 1, 32); sq += __shfl_xor(sq, 2, 32);
      const float sc = 1.f / fmaxf(sqrtf(sq), 1e-12f);
      for (int jj = 0; jj < 16; ++jj) s.k[row4][j0 + jj] *= sc;
    }
    // ---- q from last token of the whole sequence ----
    if (ch == NCHUNK - 1 && tid < 64) {
      float acc = 0.f;
      for (int j = 0; j < 64; ++j) acc += s.h[63][j] * Wq[tid * 64 + j];
      s.q[tid] = acc;
    }
    __syncthreads();

    // ---- delta-rule scan: 64 sequential rank-1 updates, M in registers,
    //      zero barriers (only cross-thread dep is the 4-lane row reduction) ----
    for (int t = 0; t < CHUNK; ++t) {
      const v4f* kp = (const v4f*)&s.k[t][j0];
      v4f k0 = kp[0], k1 = kp[1], k2 = kp[2], k3 = kp[3];
      float kseg[16];
      for (int jj = 0; jj < 4; ++jj) {
        kseg[jj]      = k0[jj];
        kseg[jj + 4]  = k1[jj];
        kseg[jj + 8]  = k2[jj];
        kseg[jj + 12] = k3[jj];
      }
      float partial = 0.f;
      for (int jj = 0; jj < 16; ++jj) partial += mreg[jj] * kseg[jj];
      partial += __shfl_xor(partial, 1, 32);
      partial += __shfl_xor(partial, 2, 32);
      const float d = s.v[t][row4] - partial;
      for (int jj = 0; jj < 16; ++jj) mreg[jj] += d * kseg[jj];
    }
  }

  // ---- dump M to LDS once for the cross-thread readout ----
  for (int jj = 0; jj < 16; ++jj) s.m[row4][j0 + jj] = mreg[jj];
  __syncthreads();

  // ---- readout: scores = (M^T q)/8, softmax, ctx = M @ attn, logits ----
  if (tid < 64) {
    float acc = 0.f;
    for (int j = 0; j < 64; ++j) acc += s.m[j][tid] * s.q[j];
    s.scores[tid] = acc * 0.125f;
  }
  __syncthreads();
  if (tid == 0) {
    float mx = s.scores[0];
    for (int i = 1; i < 64; ++i) mx = fmaxf(mx, s.scores[i]);
    float sum = 0.f;
    for (int i = 0; i < 64; ++i) sum += __expf(s.scores[i] - mx);
    s.soft[0] = mx; s.soft[1] = 1.f / sum;
  }
  __syncthreads();
  if (tid < 64) s.scores[tid] = __expf(s.scores[tid] - s.soft[0]) * s.soft[1];
  __syncthreads();
  if (tid < 64) {
    float acc = 0.f;
    for (int i = 0; i < 64; ++i) acc += s.scores[i] * s.m[tid][i];
    s.ctx[tid] = fmaxf(acc, 0.f);
  }
  __syncthreads();
  if (tid < 64) {
    float acc = s.bo[tid];
    for (int j = 0; j < 64; ++j) acc += s.ctx[j] * Wout[tid * 64 + j];
    out[b * 64 + tid] = acc;
  }
}

extern "C" void kernel_launch(void* const* d_in, const int* in_sizes, int n_in,
                              void* d_out, int out_size, void* d_ws, size_t ws_size,
                              hipStream_t stream) {
  (void)in_sizes; (void)n_in; (void)d_ws; (void)ws_size; (void)out_size;
  const int*   x     = (const int*)  d_in[0];
  const float* embed = (const float*)d_in[1];
  const float* w1    = (const float*)d_in[2];
  const float* b1    = (const float*)d_in[3];
  const float* w2    = (const float*)d_in[4];
  const float* b2    = (const float*)d_in[5];
  const float* ln_g  = (const float*)d_in[6];
  const float* ln_b  = (const float*)d_in[7];
  const float* Wk    = (const float*)d_in[8];
  const float* Wv    = (const float*)d_in[9];
  const float* Wq    = (const float*)d_in[10];
  const float* Wout  = (const float*)d_in[11];
  const float* bout  = (const float*)d_in[12];
  float* out = (float*)d_out;

  retro_fused_kernel<<<128, 256, 0, stream>>>(x, embed, w1, b1, w2, b2, ln_g, ln_b,
                                              Wk, Wv, Wq, Wout, bout, out);
}